// PolyFeatures_42588895707951
// MI455X (gfx1250) — compile-verified
//
#include <hip/hip_runtime.h>

// CDNA5 / gfx1250: wave32; V_WMMA_F32_16X16X4_F32 builds 16x16 outer-product tiles.
// Stores are emitted via inline asm in saddr form (SGPR base + 32-bit VGPR byte offset)
// to keep address math as a pure 32-bit recurrence.
typedef __attribute__((ext_vector_type(2))) float v2f;  // A/B: 64 f32 / 32 lanes = 2 VGPRs
typedef __attribute__((ext_vector_type(8))) float v8f;  // C/D: 256 f32 / 32 lanes = 8 VGPRs

#define DIM   96
#define OUTW  4753   // 1 + 96 + 96*97/2

// non-temporal f32 store: base (SGPR pair) + 32-bit byte offset (VGPR)
#define NT_STORE_B32(offB, val, base)                                   \
    asm volatile("global_store_b32 %0, %1, %2 th:TH_STORE_NT"           \
                 :: "v"(offB), "v"(val), "s"(base) : "memory")

// One wave32 per batch row.
// quad offset: off(i,j) = 97 + T(i) + (j - i),  T(i) = i*(193-i)/2
// incremental along i (fixed j): off(i+1,j) - off(i,j) = 95 - i
__global__ __launch_bounds__(256) void PolyFeatures_kernel(const float* __restrict__ z,
                                                           float* __restrict__ out,
                                                           int nrows) {
    const int lane = threadIdx.x & 31;
    const int row  = blockIdx.x * (blockDim.x >> 5) + (threadIdx.x >> 5);
    if (row >= nrows) return;  // wave-uniform; EXEC stays all-ones (WMMA requirement)

    // Wave-uniform byte offsets forced into SGPRs -> scalar bases
    const unsigned zByte   = __builtin_amdgcn_readfirstlane((unsigned)row * (DIM  * 4u));
    const unsigned outByte = __builtin_amdgcn_readfirstlane((unsigned)row * (OUTW * 4u));
    const float* __restrict__ zr   = (const float*)((const char*)z + zByte);
    float*       __restrict__ orow = (float*)((char*)out + outByte);

    const int l16 = lane & 15;

    // 6 segments of 16 values; lanes 0-15 / 16-31 hold identical copies (L1-deduped 64B loads)
    float seg[6];
#pragma unroll
    for (int s = 0; s < 6; ++s) seg[s] = zr[s * 16 + l16];

    // linear terms (coalesced 128B stores)
#pragma unroll
    for (int s = 0; s < 3; ++s) {
        const unsigned ob = (unsigned)(1 + s * 32 + lane) * 4u;
        NT_STORE_B32(ob, zr[s * 32 + lane], orow);
    }

    const int  mHi     = (lane >> 4) * 8;  // D layout: lanes 16-31 carry M = r + 8
    const bool lowHalf = (lane < 16);
    const int  rel     = l16 - mHi;        // diagonal tile: element r valid iff rel >= r

#pragma unroll
    for (int bi = 0; bi < 6; ++bi) {
        const int i0 = bi * 16 + mHi;
        // byte offset of (i0, j=bj*16+l16); one mul per tile-row
        const unsigned baseB = (unsigned)(97 + (i0 * (193 - i0)) / 2 - i0 + l16) * 4u;
        // A[m][k] = z[bi*16+m] for ALL k (robust to K-slot placement)
        v2f a; a.x = seg[bi]; a.y = seg[bi];

#pragma unroll
        for (int bj = bi; bj < 6; ++bj) {
            // B: only (VGPR0, lanes 0-15) nonzero -> single K row = z[bj*16 + n]
            v2f b; b.x = lowHalf ? seg[bj] : 0.0f; b.y = 0.0f;
            v8f c = {};
            c = __builtin_amdgcn_wmma_f32_16x16x4_f32(
                    false, a, false, b, (short)0, c, false, false);

            unsigned offB  = baseB + (unsigned)(bj * 64);  // r=0 byte offset
            int      stepB = (95 - i0) * 4;                // byte delta r -> r+1

            if (bj > bi) {
                // full tile: unconditional stores, 32-bit recurrence only
#pragma unroll
                for (int r = 0; r < 8; ++r) {
                    NT_STORE_B32(offB, c[r], orow);
                    offB += (unsigned)stepB; stepB -= 4;
                }
            } else {
                // diagonal tile: invalid lanes (j < i) redirect to byte 0 of the row;
                // orow[0] gets its true value from the FINAL store below
                // (same-wave stores to one address complete in order).
#pragma unroll
                for (int r = 0; r < 8; ++r) {
                    const unsigned o = (rel >= r) ? offB : 0u;
                    NT_STORE_B32(o, c[r], orow);
                    offB += (unsigned)stepB; stepB -= 4;
                }
            }
        }
    }

    // constant term, written LAST so it overwrites any diagonal-tile dump writes.
    // All lanes store the same value to the same address (race-free by value).
    {
        const unsigned zeroOff = 0u;
        const float one = 1.0f;
        NT_STORE_B32(zeroOff, one, orow);
    }
}

extern "C" void kernel_launch(void* const* d_in, const int* in_sizes, int n_in,
                              void* d_out, int out_size, void* d_ws, size_t ws_size,
                              hipStream_t stream) {
    const float* z   = (const float*)d_in[0];
    float*       out = (float*)d_out;
    const int nrows = in_sizes[0] / DIM;            // 32768
    const int wavesPerBlock = 8;                    // 256 threads = 8 wave32
    const int blocks = (nrows + wavesPerBlock - 1) / wavesPerBlock;
    PolyFeatures_kernel<<<blocks, 256, 0, stream>>>(z, out, nrows);
}